// structural_Des_14267881357847
// MI455X (gfx1250) — compile-verified
//
#include <hip/hip_runtime.h>
#include <hip/hip_bf16.h>

typedef float v2f __attribute__((ext_vector_type(2)));
typedef float v8f __attribute__((ext_vector_type(8)));

#define NROWS 200000
#define ROWS_PER_BLK 64
#define NCH 131
#define KPAD 132
#define NPAD 144
#define LDF 133
#define W1P_OFF 576
#define WELEMS (NPAD * KPAD) /* 19008 */

__device__ __forceinline__ v8f wmma4(v2f a, v2f b, v8f c) {
  // V_WMMA_F32_16X16X4_F32: D = A(16x4) * B(4x16) + C, fp32 throughout
  return __builtin_amdgcn_wmma_f32_16x16x4_f32(false, a, false, b, (short)0, c,
                                               false, false);
}

// C += A(16 x 4*ksteps from sA, row stride lda) * B, where B[k][n] = gB[n*ldb + k].
// rm = lane&15 serves as M for the A fragment and N for the B fragment;
// kk = (lane>>4)*2 selects the K pair per the ISA 16x4 f32 layout.
__device__ __forceinline__ v8f wmma_tile(const float* __restrict__ sA, int lda,
                                         const float* __restrict__ gB, int ldb,
                                         int ksteps, v8f c, int rm, int kk) {
  for (int ks = 0; ks < ksteps; ++ks) {
    int k0 = ks * 4 + kk;
    v2f a, b;
    a.x = sA[rm * lda + k0];
    a.y = sA[rm * lda + k0 + 1];
    b.x = gB[rm * ldb + k0];
    b.y = gB[rm * ldb + k0 + 1];
    c = wmma4(a, b, c);
  }
  return c;
}

// ---- prep: zero BN accumulators, zero-pad c1/c2 weights to [144][132] ----
__global__ void prep_kernel(const float* __restrict__ c1_w,
                            const float* __restrict__ c2_w,
                            float* __restrict__ ws) {
  int idx = blockIdx.x * 256 + threadIdx.x;
  int tot = W1P_OFF + 2 * WELEMS;
  if (idx >= tot) return;
  if (idx < W1P_OFF) { ws[idx] = 0.0f; return; }
  int j = idx - W1P_OFF;
  const float* src = (j < WELEMS) ? c1_w : c2_w;
  int r = (j < WELEMS) ? j : j - WELEMS;
  int n = r / KPAD, k = r - n * KPAD;
  ws[idx] = (n < NCH && k < NCH) ? src[n * NCH + k] : 0.0f;
}

// ---- kernel A: features + c1 GEMM (WMMA) + BN partial sums ----
__global__ void __launch_bounds__(256) feat_c1_kernel(
    const float* __restrict__ corner, const float* __restrict__ normal,
    const int* __restrict__ neighbour,
    const float* __restrict__ conv_w, const float* __restrict__ conv_b,
    const float* __restrict__ l3_w, const float* __restrict__ l3_b,
    const float* __restrict__ l4_w, const float* __restrict__ l4_b,
    const float* __restrict__ theta, const float* __restrict__ phi,
    const float* __restrict__ c1_b, const float* __restrict__ W1p,
    float* __restrict__ gsum, float* __restrict__ gssq,
    float* __restrict__ yout) {
  __shared__ float  s_corner[ROWS_PER_BLK * 9];
  __shared__ float  s_f[ROWS_PER_BLK * 16];
  __shared__ float4 s_kern[256];
  __shared__ float  s_frc[ROWS_PER_BLK * 33];
  __shared__ float  s_h[ROWS_PER_BLK * 65];
  __shared__ float  s_feat[ROWS_PER_BLK * LDF];
  __shared__ float  s_sum[NCH];
  __shared__ float  s_ssq[NCH];

  const int tid = threadIdx.x;
  const int lane = tid & 31;
  const int wave = tid >> 5;
  const int rowBase = blockIdx.x * ROWS_PER_BLK;

  // stage 0: stage inputs into LDS
  for (int i = tid; i < ROWS_PER_BLK * 9; i += 256)
    s_corner[i] = corner[rowBase * 9 + i];
  {
    float th = theta[tid], ph = phi[tid];
    float st = __sinf(th), ct = __cosf(th);
    float sp = __sinf(ph), cp = __cosf(ph);
    float kx = st * sp, ky = st * cp, kz = ct;
    s_kern[tid] = make_float4(kx, ky, kz, kx * kx + ky * ky + kz * kz);
  }
  if (tid < ROWS_PER_BLK) {
    int row = rowBase + tid;
    float nx = normal[row * 3 + 0], ny = normal[row * 3 + 1], nz = normal[row * 3 + 2];
    s_feat[tid * LDF + 0] = nx; s_feat[tid * LDF + 1] = ny; s_feat[tid * LDF + 2] = nz;
    s_feat[tid * LDF + 131] = 0.0f;  // K padding for c1
    s_f[tid * 16 + 0] = nx; s_f[tid * 16 + 1] = ny;
    s_f[tid * 16 + 2] = nz; s_f[tid * 16 + 3] = nx * nx + ny * ny + nz * nz;
    for (int a = 1; a < 4; ++a) {
      int j = neighbour[row * 3 + (a - 1)];
      float x = normal[j * 3 + 0], y = normal[j * 3 + 1], z = normal[j * 3 + 2];
      s_f[tid * 16 + a * 4 + 0] = x; s_f[tid * 16 + a * 4 + 1] = y;
      s_f[tid * 16 + a * 4 + 2] = z; s_f[tid * 16 + a * 4 + 3] = x * x + y * y + z * z;
    }
  }
  if (tid < NCH) { s_sum[tid] = 0.0f; s_ssq[tid] = 0.0f; }
  __syncthreads();

  // stage 1: Face_Rotate_Conv. mean over 3 windows == dot with window-summed arr.
  {
    int row = tid >> 2, q = tid & 3;
    float arr[12];
    for (int j = 0; j < 9; ++j) arr[j] = s_corner[row * 9 + j];
    arr[9] = arr[0]; arr[10] = arr[1]; arr[11] = arr[2];
    float s6[6];
    for (int k = 0; k < 6; ++k) s6[k] = arr[k] + arr[k + 3] + arr[k + 6];
    for (int oo = 0; oo < 8; ++oo) {
      int o = q * 8 + oo;
      float acc = 0.0f;
      for (int k = 0; k < 6; ++k) acc += conv_w[o * 6 + k] * s6[k];
      s_frc[row * 33 + o] = acc * (1.0f / 3.0f) + conv_b[o];
    }
  }

  // stage 2: kernel correlation (16 exps per m, 16 m's per thread)
  {
    int row = tid >> 2, q = tid & 3;
    float fx[4], fy[4], fz[4], f2[4];
    for (int a = 0; a < 4; ++a) {
      fx[a] = s_f[row * 16 + a * 4 + 0];
      fy[a] = s_f[row * 16 + a * 4 + 1];
      fz[a] = s_f[row * 16 + a * 4 + 2];
      f2[a] = s_f[row * 16 + a * 4 + 3];
    }
    for (int mm = 0; mm < 16; ++mm) {
      int m = q * 16 + mm;
      float acc = 0.0f;
      for (int b = 0; b < 4; ++b) {
        float4 kb = s_kern[m * 4 + b];
        for (int a = 0; a < 4; ++a) {
          float d = f2[a] + kb.w - 2.0f * (fx[a] * kb.x + fy[a] * kb.y + fz[a] * kb.z);
          acc += __expf(d * -12.5f);  // 1/(-2*sigma^2), sigma=0.2
        }
      }
      s_feat[row * LDF + 67 + m] = acc * (1.0f / 16.0f);
    }
  }
  __syncthreads();

  // stage 3: l3 = relu(frc @ W3^T + b3), M=64 K=32 N=64, WMMA
  {
    int rm = lane & 15, kk = (lane >> 4) << 1, rh = (lane >> 4) << 3;
    for (int t = wave; t < 16; t += 8) {
      int mt = t & 3, nt = t >> 2;
      v8f c;
      float bv = l3_b[nt * 16 + rm];
      for (int r = 0; r < 8; ++r) c[r] = bv;
      c = wmma_tile(s_frc + mt * 16 * 33, 33, l3_w + nt * 16 * 32, 32, 8, c, rm, kk);
      for (int r = 0; r < 8; ++r)
        s_h[(mt * 16 + rh + r) * 65 + nt * 16 + rm] = fmaxf(c[r], 0.0f);
    }
  }
  __syncthreads();

  // stage 4: x = h @ W4^T + b4, M=64 K=64 N=64, WMMA; x -> feat[3..66]
  {
    int rm = lane & 15, kk = (lane >> 4) << 1, rh = (lane >> 4) << 3;
    for (int t = wave; t < 16; t += 8) {
      int mt = t & 3, nt = t >> 2;
      v8f c;
      float bv = l4_b[nt * 16 + rm];
      for (int r = 0; r < 8; ++r) c[r] = bv;
      c = wmma_tile(s_h + mt * 16 * 65, 65, l4_w + nt * 16 * 64, 64, 16, c, rm, kk);
      for (int r = 0; r < 8; ++r)
        s_feat[(mt * 16 + rh + r) * LDF + 3 + nt * 16 + rm] = c[r];
    }
  }
  __syncthreads();

  // stage 5: y = feat @ W1^T + b1, M=64 K=132 N=144(131), WMMA; y -> d_out, BN sums
  {
    int rm = lane & 15, kk = (lane >> 4) << 1, rh = (lane >> 4) << 3;
    for (int t = wave; t < 36; t += 8) {
      int mt = t & 3, nt = t >> 2;
      int col = nt * 16 + rm;
      v8f c;
      float bv = (col < NCH) ? c1_b[col] : 0.0f;
      for (int r = 0; r < 8; ++r) c[r] = bv;
      c = wmma_tile(s_feat + mt * 16 * LDF, LDF, W1p + nt * 16 * KPAD, KPAD, 33, c, rm, kk);
      if (col < NCH) {
        for (int r = 0; r < 8; ++r) {
          int row = rowBase + mt * 16 + rh + r;
          float v = c[r];
          yout[row * NCH + col] = v;
          atomicAdd(&s_sum[col], v);
          atomicAdd(&s_ssq[col], v * v);
        }
      }
    }
  }
  __syncthreads();
  if (tid < NCH) {
    atomicAdd(&gsum[tid], s_sum[tid]);
    atomicAdd(&gssq[tid], s_ssq[tid]);
  }
}

// ---- kernel B: fold BN into per-channel scale/shift ----
__global__ void bn_stats_kernel(const float* __restrict__ gsum,
                                const float* __restrict__ gssq,
                                const float* __restrict__ bn_g,
                                const float* __restrict__ bn_b,
                                float* __restrict__ scale,
                                float* __restrict__ shift) {
  int tid = threadIdx.x;
  if (tid < NCH) {
    float inv_n = 1.0f / (float)NROWS;
    float mu = gsum[tid] * inv_n;
    float var = gssq[tid] * inv_n - mu * mu;
    float a = bn_g[tid] * rsqrtf(var + 1e-5f);
    scale[tid] = a;
    shift[tid] = bn_b[tid] - mu * a;
  }
}

// ---- kernel C: out = relu(y*scale+shift) @ W2^T + b2, in place over d_out ----
__global__ void __launch_bounds__(256) bn_c2_kernel(
    const float* __restrict__ scale, const float* __restrict__ shift,
    const float* __restrict__ W2p, const float* __restrict__ c2_b,
    float* __restrict__ out) {
  __shared__ float s_t[ROWS_PER_BLK * LDF];
  const int tid = threadIdx.x;
  const int lane = tid & 31, wave = tid >> 5;
  const int rowBase = blockIdx.x * ROWS_PER_BLK;

  for (int i = tid; i < ROWS_PER_BLK * NCH; i += 256) {
    int r = i / NCH, c = i - r * NCH;
    float v = out[(rowBase + r) * NCH + c];
    s_t[r * LDF + c] = fmaxf(v * scale[c] + shift[c], 0.0f);
  }
  if (tid < ROWS_PER_BLK) s_t[tid * LDF + 131] = 0.0f;
  __syncthreads();

  int rm = lane & 15, kk = (lane >> 4) << 1, rh = (lane >> 4) << 3;
  for (int t = wave; t < 36; t += 8) {
    int mt = t & 3, nt = t >> 2;
    int col = nt * 16 + rm;
    v8f c;
    float bv = (col < NCH) ? c2_b[col] : 0.0f;
    for (int r = 0; r < 8; ++r) c[r] = bv;
    c = wmma_tile(s_t + mt * 16 * LDF, LDF, W2p + nt * 16 * KPAD, KPAD, 33, c, rm, kk);
    if (col < NCH)
      for (int r = 0; r < 8; ++r)
        out[(rowBase + mt * 16 + rh + r) * NCH + col] = c[r];
  }
}

extern "C" void kernel_launch(void* const* d_in, const int* in_sizes, int n_in,
                              void* d_out, int out_size, void* d_ws, size_t ws_size,
                              hipStream_t stream) {
  (void)in_sizes; (void)n_in; (void)out_size; (void)ws_size;
  const float* corner    = (const float*)d_in[0];
  const float* normal    = (const float*)d_in[1];
  const int*   neighbour = (const int*)d_in[2];
  const float* conv_w    = (const float*)d_in[3];
  const float* conv_b    = (const float*)d_in[4];
  const float* l3_w      = (const float*)d_in[5];
  const float* l3_b      = (const float*)d_in[6];
  const float* l4_w      = (const float*)d_in[7];
  const float* l4_b      = (const float*)d_in[8];
  const float* theta     = (const float*)d_in[9];
  const float* phi       = (const float*)d_in[10];
  const float* c1_w      = (const float*)d_in[11];
  const float* c1_b      = (const float*)d_in[12];
  const float* c2_w      = (const float*)d_in[13];
  const float* c2_b      = (const float*)d_in[14];
  const float* bn_g      = (const float*)d_in[15];
  const float* bn_b      = (const float*)d_in[16];
  float* out = (float*)d_out;
  float* ws  = (float*)d_ws;

  float* gsum  = ws + 0;
  float* gssq  = ws + NCH;
  float* scale = ws + 2 * NCH;
  float* shift = ws + 3 * NCH;
  float* W1p   = ws + W1P_OFF;
  float* W2p   = W1p + WELEMS;

  int tot = W1P_OFF + 2 * WELEMS;
  prep_kernel<<<(tot + 255) / 256, 256, 0, stream>>>(c1_w, c2_w, ws);

  int nblk = NROWS / ROWS_PER_BLK;  // 3125, exact
  feat_c1_kernel<<<nblk, 256, 0, stream>>>(corner, normal, neighbour, conv_w,
                                           conv_b, l3_w, l3_b, l4_w, l4_b,
                                           theta, phi, c1_b, W1p, gsum, gssq, out);
  bn_stats_kernel<<<1, 256, 0, stream>>>(gsum, gssq, bn_g, bn_b, scale, shift);
  bn_c2_kernel<<<nblk, 256, 0, stream>>>(scale, shift, W2p, c2_b, out);
}